// GHMLoss_25056839205987
// MI455X (gfx1250) — compile-verified
//
#include <hip/hip_runtime.h>
#include <hip/hip_bf16.h>
#include <math.h>

#define BINS 10
#define TPB  256
#define MAXB 2048

// ---- CDNA5 async global->LDS path (gfx1250), with safe fallback -----------
#if defined(__HIP_DEVICE_COMPILE__) && __has_builtin(__builtin_amdgcn_global_load_async_to_lds_b128)
#define USE_ASYNC 1
#else
#define USE_ASYNC 0
#endif

#if defined(__HIP_DEVICE_COMPILE__) && __has_builtin(__builtin_amdgcn_s_wait_asynccnt)
#define WAIT_ASYNC(n) __builtin_amdgcn_s_wait_asynccnt(n)
#else
#define WAIT_ASYNC(n) asm volatile("s_wait_asynccnt " #n)
#endif

// Builtin parameter types (from hipcc diagnostic): int4-vector pointers,
// global (AS1) source and LDS (AS3) destination.
typedef int v4i __attribute__((vector_size(4 * sizeof(int))));
typedef __attribute__((address_space(1))) v4i as1_v4i;
typedef __attribute__((address_space(3))) v4i as3_v4i;

// ---------------------------------------------------------------------------
// Pass 1: fused sigmoid/BCE + 10-bin histogram of valid counts AND per-bin
// loss sums in a single stream over the data. Per-thread LDS slots (no
// atomics). Each block emits 10 counts + 10 sums to workspace.
// ---------------------------------------------------------------------------
__global__ __launch_bounds__(TPB) void ghm_pass1(
    const float* __restrict__ in, const float* __restrict__ tg,
    const float* __restrict__ lw, float* __restrict__ ws_cnt,
    float* __restrict__ ws_sum, int n, int nTiles) {
  __shared__ float s_cnt[BINS * TPB];
  __shared__ float s_sum[BINS * TPB];
#if USE_ASYNC
  __shared__ float4 stg_in[2][TPB];
  __shared__ float4 stg_tg[2][TPB];
  __shared__ float4 stg_lw[2][TPB];
#endif
  const int tid = threadIdx.x;

#pragma unroll
  for (int b = 0; b < BINS; ++b) {
    s_cnt[b * TPB + tid] = 0.0f;
    s_sum[b * TPB + tid] = 0.0f;
  }

  const float4* in4 = (const float4*)in;
  const float4* tg4 = (const float4*)tg;
  const float4* lw4 = (const float4*)lw;

  auto accum = [&](float x, float t, float w) {
    // stable shared-exp form: e = exp(-|x|)
    float e = __expf(-fabsf(x));
    float r = __fdividef(1.0f, 1.0f + e);            // 1/(1+e)
    float prob = (x >= 0.0f) ? r : (1.0f - r);       // sigmoid(x)
    float g = fabsf(prob - t);
    int bi = (int)(g * (float)BINS);                 // g >= 0 -> floor
    bi = bi > (BINS - 1) ? (BINS - 1) : bi;
    // softplus(x) - x*t, stable: max(x,0) + log1p(e) - x*t
    float loss = fmaxf(x, 0.0f) + __logf(1.0f + e) - x * t;
    float v = (w > 0.0f) ? 1.0f : 0.0f;
    s_cnt[bi * TPB + tid] += v;
    s_sum[bi * TPB + tid] += v * loss;
  };

#if USE_ASYNC
  auto issue = [&](int tile, int buf) {
    int gi = tile * TPB + tid;  // float4 index
    __builtin_amdgcn_global_load_async_to_lds_b128(
        (as1_v4i*)(in4 + gi), (as3_v4i*)&stg_in[buf][tid], 0, 0);
    __builtin_amdgcn_global_load_async_to_lds_b128(
        (as1_v4i*)(tg4 + gi), (as3_v4i*)&stg_tg[buf][tid], 0, 0);
    __builtin_amdgcn_global_load_async_to_lds_b128(
        (as1_v4i*)(lw4 + gi), (as3_v4i*)&stg_lw[buf][tid], 0, 0);
  };
  int buf = 0;
  int t = (int)blockIdx.x;
  if (t < nTiles) issue(t, buf);
  for (; t < nTiles; t += (int)gridDim.x) {
    int nxt = t + (int)gridDim.x;
    if (nxt < nTiles) {
      issue(nxt, buf ^ 1);   // keep 6 in flight
      WAIT_ASYNC(3);         // in-order completion: current tile's 3 are done
    } else {
      WAIT_ASYNC(0);         // drain
    }
    float4 a = stg_in[buf][tid];
    float4 b = stg_tg[buf][tid];
    float4 c = stg_lw[buf][tid];
    accum(a.x, b.x, c.x);
    accum(a.y, b.y, c.y);
    accum(a.z, b.z, c.z);
    accum(a.w, b.w, c.w);
    buf ^= 1;
  }
#else
  for (int t = (int)blockIdx.x; t < nTiles; t += (int)gridDim.x) {
    int gi = t * TPB + tid;
    float4 a = in4[gi];
    float4 b = tg4[gi];
    float4 c = lw4[gi];
    accum(a.x, b.x, c.x);
    accum(a.y, b.y, c.y);
    accum(a.z, b.z, c.z);
    accum(a.w, b.w, c.w);
  }
#endif

  // tail elements (n not a multiple of 1024) handled by block 0
  int done = nTiles * TPB * 4;
  if (blockIdx.x == 0) {
    for (int i = done + tid; i < n; i += TPB) accum(in[i], tg[i], lw[i]);
  }

  __syncthreads();
  for (int off = TPB >> 1; off > 0; off >>= 1) {
    if (tid < off) {
#pragma unroll
      for (int b = 0; b < BINS; ++b) {
        s_cnt[b * TPB + tid] += s_cnt[b * TPB + tid + off];
        s_sum[b * TPB + tid] += s_sum[b * TPB + tid + off];
      }
    }
    __syncthreads();
  }
  if (tid < BINS) {
    ws_cnt[tid * gridDim.x + blockIdx.x] = s_cnt[tid * TPB];
    ws_sum[tid * gridDim.x + blockIdx.x] = s_sum[tid * TPB];
  }
}

// ---------------------------------------------------------------------------
// Pass 2: one wave32 per bin reduces the per-block partials in a fixed
// (deterministic) order; thread 0 applies the GHM weighting formula.
// ---------------------------------------------------------------------------
__global__ __launch_bounds__(32 * BINS) void ghm_pass2(
    const float* __restrict__ ws_cnt, const float* __restrict__ ws_sum,
    float* __restrict__ out, int nb) {
  __shared__ float c[BINS];
  __shared__ float s[BINS];
  int wave = threadIdx.x >> 5;
  int lane = threadIdx.x & 31;
  float cc = 0.0f, ss = 0.0f;
  for (int i = lane; i < nb; i += 32) {
    cc += ws_cnt[wave * nb + i];
    ss += ws_sum[wave * nb + i];
  }
#pragma unroll
  for (int o = 16; o > 0; o >>= 1) {
    cc += __shfl_down(cc, o);
    ss += __shfl_down(ss, o);
  }
  if (lane == 0) {
    c[wave] = cc;
    s[wave] = ss;
  }
  __syncthreads();
  if (threadIdx.x == 0) {
    float tot = 0.0f;
    for (int b = 0; b < BINS; ++b) tot += c[b];
    tot = fmaxf(tot, 1.0f);
    float nbins = 0.0f, acc = 0.0f;
    for (int b = 0; b < BINS; ++b) {
      if (c[b] > 0.0f) {
        nbins += 1.0f;
        acc += (tot / fmaxf(c[b], 1.0f)) * s[b];  // per_bin_w[b] * loss_sum[b]
      }
    }
    out[0] = acc / fmaxf(nbins, 1.0f) / tot;
  }
}

extern "C" void kernel_launch(void* const* d_in, const int* in_sizes, int n_in,
                              void* d_out, int out_size, void* d_ws, size_t ws_size,
                              hipStream_t stream) {
  (void)n_in;
  (void)out_size;
  const float* in = (const float*)d_in[0];
  const float* tg = (const float*)d_in[1];
  const float* lw = (const float*)d_in[2];
  int n = in_sizes[0];

  int n4 = n >> 2;            // float4 count (tail handled in-kernel)
  int nTiles = n4 / TPB;      // 1024 elements per tile

  int grid = MAXB;
  size_t per_block_bytes = (size_t)2 * BINS * sizeof(float);
  if ((size_t)grid * per_block_bytes > ws_size)
    grid = (int)(ws_size / per_block_bytes);
  if (grid < 1) grid = 1;
  if (nTiles > 0 && grid > nTiles) grid = nTiles;

  float* ws_cnt = (float*)d_ws;
  float* ws_sum = ws_cnt + (size_t)BINS * grid;

  ghm_pass1<<<grid, TPB, 0, stream>>>(in, tg, lw, ws_cnt, ws_sum, n, nTiles);
  ghm_pass2<<<1, 32 * BINS, 0, stream>>>(ws_cnt, ws_sum, (float*)d_out, grid);
}